// TopKRouter_68728066670791
// MI455X (gfx1250) — compile-verified
//
#include <hip/hip_runtime.h>
#include <hip/hip_bf16.h>
#include <math.h>
#include <stdint.h>

#define HIDDEN      4096
#define NUM_EXPERTS 64
#define TOP_K       8

// ---- GEMM tiling ----------------------------------------------------------
#define BLOCK_T        128                 // 4 waves per block
#define ROWS_PER_WAVE  32                  // two 16-row WMMA tiles per wave
#define ROWS_PER_BLOCK 128                 // 4 waves * 32 rows
#define KC             64                  // K-chunk staged in LDS
#define LDS_STRIDE     (KC + 4)            // 68 floats: conflict-free b64 reads
#define NCHUNK         (HIDDEN / KC)       // 64 chunks

typedef __attribute__((ext_vector_type(2))) float v2f;
typedef __attribute__((ext_vector_type(8))) float v8f;

// Async global->LDS copy (ASYNCcnt path). LDS dest address VGPR, 64-bit
// global source address. Inline asm keeps this portable across toolchains.
__device__ __forceinline__ void async_ld_lds_b128(unsigned lds_byte, const float* gsrc) {
    asm volatile("global_load_async_to_lds_b128 %0, %1, off"
                 :: "v"(lds_byte), "v"(gsrc)
                 : "memory");
}

__device__ __forceinline__ void wait_async0() {
#if __has_builtin(__builtin_amdgcn_s_wait_asynccnt)
    __builtin_amdgcn_s_wait_asynccnt(0);
#else
    asm volatile("s_wait_asynccnt 0" ::: "memory");
#endif
}

// ---------------------------------------------------------------------------
// Kernel 1: router logits GEMM  [T,4096] x [4096,64]^T -> [T,64]
// x streams straight from HBM (pipelined global_load_b64 + prefetch);
// gate_w K-chunks are staged once per block into double-buffered LDS via
// async copies, B fragments come from ds_load_b64.
// ---------------------------------------------------------------------------
__global__ __launch_bounds__(BLOCK_T) void router_gemm_kernel(
    const float* __restrict__ x,
    const float* __restrict__ gw,       // [64, 4096] (nn.Linear layout)
    float* __restrict__ logits,         // [T, 64]
    int T)
{
    __shared__ float bsm[2][NUM_EXPERTS * LDS_STRIDE];   // 2 x 17,408 B

    const int tid  = threadIdx.x;
    const int lane = tid & 31;
    const int wave = tid >> 5;
    const int m    = lane & 15;        // row (A) / col (B) within 16x16 tile
    const int half = lane >> 4;        // 0 -> K{0,1}, 1 -> K{2,3}

    const int rowBase = blockIdx.x * ROWS_PER_BLOCK + wave * ROWS_PER_WAVE;

    const float* aPtr0 = x + (size_t)(rowBase + m) * HIDDEN + 2 * half;
    const float* aPtr1 = aPtr0 + (size_t)16 * HIDDEN;

    // Flat LDS pointer's low 32 bits == AS3 byte offset on this target.
    const unsigned smb[2] = { (unsigned)(uintptr_t)&bsm[0][0],
                              (unsigned)(uintptr_t)&bsm[1][0] };

    // Async fill of gate_w chunk [64 experts x KC] starting at column k0.
    // 64*16 float4s / 128 threads = 8 per thread; 16 consecutive lanes cover
    // one expert row => 256B contiguous per group.
    auto fill = [&](int b, int k0) {
        const unsigned base = smb[b];
        #pragma unroll
        for (int j = 0; j < 8; ++j) {
            const int idx = tid + j * BLOCK_T;     // 0..1023
            const int e   = idx >> 4;              // expert row 0..63
            const int c4  = idx & 15;              // float4 column 0..15
            async_ld_lds_b128(base + (unsigned)((e * LDS_STRIDE + 4 * c4) * 4),
                              gw + (size_t)e * HIDDEN + k0 + 4 * c4);
        }
    };

    fill(0, 0);

    v8f acc[2][4];
    #pragma unroll
    for (int mt = 0; mt < 2; ++mt)
        #pragma unroll
        for (int t = 0; t < 4; ++t)
            acc[mt][t] = (v8f){};

    wait_async0();
    __syncthreads();

    for (int c = 0; c < NCHUNK; ++c) {
        if (c + 1 < NCHUNK) fill((c + 1) & 1, (c + 1) * KC);   // overlap fill

        const float* bb = &bsm[c & 1][0];
        const int kb = c * KC;

        #pragma unroll 4
        for (int kc = 0; kc < KC; kc += 4) {
            const int k = kb + kc;
            __builtin_prefetch(aPtr0 + k + 1024, 0, 1);        // stream-ahead A
            __builtin_prefetch(aPtr1 + k + 1024, 0, 1);

            v2f a0 = *(const v2f*)(aPtr0 + k);
            v2f a1 = *(const v2f*)(aPtr1 + k);

            #pragma unroll
            for (int t = 0; t < 4; ++t) {
                // conflict-free: lane(half,m) hits banks {4m+2*half, +1}
                v2f bw = *(const v2f*)(bb + (t * 16 + m) * LDS_STRIDE + kc + 2 * half);
                acc[0][t] = __builtin_amdgcn_wmma_f32_16x16x4_f32(
                    false, a0, false, bw, (short)0, acc[0][t], false, false);
                acc[1][t] = __builtin_amdgcn_wmma_f32_16x16x4_f32(
                    false, a1, false, bw, (short)0, acc[1][t], false, false);
            }
        }

        wait_async0();      // next buffer filled
        __syncthreads();    // all waves done with current buffer (WAR close)
    }

    // C/D layout: VGPR v holds (M = v + 8*half, N = m) within each tile.
    #pragma unroll
    for (int mt = 0; mt < 2; ++mt) {
        float* out = logits + (size_t)(rowBase + mt * 16) * NUM_EXPERTS;
        #pragma unroll
        for (int v = 0; v < 8; ++v) {
            const size_t r = (size_t)(v + 8 * half) * NUM_EXPERTS;
            #pragma unroll
            for (int t = 0; t < 4; ++t)
                out[r + t * 16 + m] = acc[mt][t][v];
        }
    }
}

// ---------------------------------------------------------------------------
// Kernel 2: per-token top-8 selection + softmax over the selected logits.
// ---------------------------------------------------------------------------
__global__ __launch_bounds__(256) void topk_softmax_kernel(
    const float* __restrict__ logits,   // [T, 64]
    float* __restrict__ outIdx,         // [T, 8] (indices as float)
    float* __restrict__ outW,           // [T, 8]
    int T)
{
    const int t = blockIdx.x * blockDim.x + threadIdx.x;
    if (t >= T) return;

    const float* row = logits + (size_t)t * NUM_EXPERTS;
    float vals[NUM_EXPERTS];
    #pragma unroll
    for (int e = 0; e < NUM_EXPERTS; ++e) vals[e] = row[e];

    unsigned long long taken = 0ull;
    float tv[TOP_K];
    int   ti[TOP_K];

    #pragma unroll
    for (int kk = 0; kk < TOP_K; ++kk) {
        float best = -INFINITY;
        int   bi   = 0;
        #pragma unroll
        for (int e = 0; e < NUM_EXPERTS; ++e) {
            const bool used = (taken >> e) & 1ull;
            if (!used && vals[e] > best) { best = vals[e]; bi = e; }
        }
        taken |= (1ull << bi);
        tv[kk] = best;
        ti[kk] = bi;
    }

    const float mx = tv[0];             // picks are descending
    float ex[TOP_K];
    float s = 0.0f;
    #pragma unroll
    for (int kk = 0; kk < TOP_K; ++kk) { ex[kk] = __expf(tv[kk] - mx); s += ex[kk]; }
    const float inv = 1.0f / s;

    #pragma unroll
    for (int kk = 0; kk < TOP_K; ++kk) {
        outIdx[(size_t)t * TOP_K + kk] = (float)ti[kk];
        outW  [(size_t)t * TOP_K + kk] = ex[kk] * inv;
    }
}

// ---------------------------------------------------------------------------
extern "C" void kernel_launch(void* const* d_in, const int* in_sizes, int n_in,
                              void* d_out, int out_size, void* d_ws, size_t ws_size,
                              hipStream_t stream) {
    const float* x  = (const float*)d_in[0];   // [2, 8192, 4096] f32
    const float* gw = (const float*)d_in[1];   // [64, 4096] f32
    float* out = (float*)d_out;

    const int T = in_sizes[0] / HIDDEN;        // 16384 tokens

    float* logits = out;                                   // T*64
    float* outIdx = out + (size_t)T * NUM_EXPERTS;         // T*8
    float* outW   = outIdx + (size_t)T * TOP_K;            // T*8

    router_gemm_kernel<<<dim3(T / ROWS_PER_BLOCK), dim3(BLOCK_T), 0, stream>>>(x, gw, logits, T);
    topk_softmax_kernel<<<dim3((T + 255) / 256), dim3(256), 0, stream>>>(logits, outIdx, outW, T);
}